// ConsistencyLoss_14053132992786
// MI455X (gfx1250) — compile-verified
//
#include <hip/hip_runtime.h>
#include <hip/hip_bf16.h>

typedef __attribute__((ext_vector_type(2))) float v2f;
typedef __attribute__((ext_vector_type(8))) float v8f;

#define BIMG 8
#define CCH 64
#define HW 65536          // 256*256 pixels per image
#define FHW 4096          // 64*64 low-res feature map
#define S_SEG 256         // superpixels per image

// workspace layout (in floats)
#define OFF_FEATMEAN 0                       // BIMG*FHW = 32768
#define OFF_SUM1 (BIMG * FHW)                // BIMG*S_SEG
#define OFF_SUM2 (OFF_SUM1 + BIMG * S_SEG)
#define OFF_CNT  (OFF_SUM2 + BIMG * S_SEG)
#define OFF_M1   (OFF_CNT  + BIMG * S_SEG)
#define OFF_M2   (OFF_M1   + BIMG * S_SEG)

// ---------------------------------------------------------------- K0: zero
__global__ void k_zero(float* __restrict__ ws, float* __restrict__ out) {
    int i = blockIdx.x * blockDim.x + threadIdx.x;
    if (i < 3 * BIMG * S_SEG) ws[OFF_SUM1 + i] = 0.0f;
    if (i == 0) out[0] = 0.0f;
}

// ------------------------------------------- K1: channel mean of feature map
__global__ void k_featmean(const float* __restrict__ feat, float* __restrict__ fm) {
    int i = blockIdx.x * blockDim.x + threadIdx.x;   // 0 .. BIMG*FHW-1
    int b = i >> 12;
    int yx = i & (FHW - 1);
    const float* p = feat + ((size_t)b * CCH) * FHW + yx;
    float s = 0.0f;
#pragma unroll 8
    for (int c = 0; c < CCH; ++c) s += __builtin_nontemporal_load(p + (size_t)c * FHW);
    fm[i] = s * (1.0f / (float)CCH);
}

// --------------- K2: WMMA channel-sum of input + bilinear feat + segment acc
// grid: BIMG * 128 blocks of 256 threads. Each block covers 512 pixels:
// 8 waves * 4 chains * 16 pixels. Each chain: 16x V_WMMA_F32_16X16X4_F32 with
// A = ones, so D[:,n] = sum over 64 channels of pixel (pbase+n).
// Input is a 134 MB single-use stream -> non-temporal loads (TH=NT) so it
// doesn't rinse the reused feature-mean map / segment bins out of L2.
__global__ void __launch_bounds__(256) k_main(const float* __restrict__ in,
                                              const int* __restrict__ sp,
                                              const float* __restrict__ fmg,
                                              float* __restrict__ gsum1,
                                              float* __restrict__ gsum2,
                                              float* __restrict__ gcnt) {
    __shared__ float lsum1[S_SEG];
    __shared__ float lsum2[S_SEG];
    __shared__ float lcnt[S_SEG];

    int tid = threadIdx.x;
    lsum1[tid] = 0.0f;
    lsum2[tid] = 0.0f;
    lcnt[tid]  = 0.0f;
    __syncthreads();

    int b      = blockIdx.x >> 7;          // 128 blocks per image
    int pblock = (blockIdx.x & 127) * 512; // pixel base of this block
    int wave   = tid >> 5;
    int lane   = tid & 31;
    int n      = lane & 15;                // pixel column within tile
    int half   = lane >> 4;                // which half-wave (channel pair)

    const v2f ones = {1.0f, 1.0f};
    const float* base_b = in + (size_t)b * CCH * HW;
    const float* fm     = fmg + b * FHW;

    for (int it = 0; it < 4; ++it) {
        int pbase = pblock + (wave * 4 + it) * 16;

        v8f acc = {};
        // lane L supplies B[k, n] with n = L&15; channels kc + 2*half + {0,1}
        const float* pp = base_b + (size_t)(2 * half) * HW + pbase + n;
#pragma unroll
        for (int kc = 0; kc < CCH; kc += 4) {
            v2f bb;
            bb.x = __builtin_nontemporal_load(pp);
            bb.y = __builtin_nontemporal_load(pp + HW);
            pp += (size_t)4 * HW;
            acc = __builtin_amdgcn_wmma_f32_16x16x4_f32(
                false, ones, false, bb, (short)0, acc, false, false);
        }

        if (lane < 16) {
            float px1 = acc[0] * (1.0f / (float)CCH);   // channel mean, path 1

            int p  = pbase + n;
            int py = p >> 8;
            int px = p & 255;
            // align_corners bilinear sample of the 64x64 channel-mean map
            float fy = (float)py * (63.0f / 255.0f);
            float fx = (float)px * (63.0f / 255.0f);
            int y0 = (int)fy, x0 = (int)fx;
            int y1 = min(y0 + 1, 63), x1 = min(x0 + 1, 63);
            float wy = fy - (float)y0;
            float wx = fx - (float)x0;
            float v00 = fm[(y0 << 6) + x0];
            float v01 = fm[(y0 << 6) + x1];
            float v10 = fm[(y1 << 6) + x0];
            float v11 = fm[(y1 << 6) + x1];
            float r0  = v00 * (1.0f - wy) + v10 * wy;
            float r1  = v01 * (1.0f - wy) + v11 * wy;
            float px2 = r0 * (1.0f - wx) + r1 * wx;     // channel mean, path 2

            int s = __builtin_nontemporal_load(sp + b * HW + p);
            atomicAdd(&lsum1[s], px1);
            atomicAdd(&lsum2[s], px2);
            atomicAdd(&lcnt[s], 1.0f);
        }
    }
    __syncthreads();

    // one bin per thread -> 3 global atomics per bin per block
    float c = lcnt[tid];
    if (c != 0.0f) {
        atomicAdd(&gsum1[b * S_SEG + tid], lsum1[tid]);
        atomicAdd(&gsum2[b * S_SEG + tid], lsum2[tid]);
        atomicAdd(&gcnt[b * S_SEG + tid], c);
    }
}

// ------------------------------------------------ K3: finalize segment means
__global__ void k_means(const float* __restrict__ sum1,
                        const float* __restrict__ sum2,
                        const float* __restrict__ cnt,
                        float* __restrict__ m1, float* __restrict__ m2) {
    int i = blockIdx.x * blockDim.x + threadIdx.x;   // 0 .. BIMG*S_SEG-1
    float c = cnt[i];
    // num[b] == S always, so the validity mask is identity; cnt==0 -> 0
    m1[i] = (c > 0.0f) ? sum1[i] / c : 0.0f;
    m2[i] = (c > 0.0f) ? sum2[i] / c : 0.0f;
}

// -------------------- K4: L1 between similarity matrices, global mean to out
__global__ void __launch_bounds__(256) k_loss(const float* __restrict__ m1,
                                              const float* __restrict__ m2,
                                              float* __restrict__ out) {
    __shared__ float a1[S_SEG];
    __shared__ float a2[S_SEG];
    __shared__ float red[256];

    int b = blockIdx.x;
    int j = threadIdx.x;
    a1[j] = m1[b * S_SEG + j];
    a2[j] = m2[b * S_SEG + j];
    __syncthreads();

    float x1 = a1[j], x2 = a2[j];
    float acc = 0.0f;
#pragma unroll 4
    for (int k = 0; k < S_SEG; ++k) {
        acc += fabsf(fabsf(x1 - a1[k]) - fabsf(x2 - a2[k]));
    }
    red[j] = acc;
    __syncthreads();
    for (int off = 128; off > 0; off >>= 1) {
        if (j < off) red[j] += red[j + off];
        __syncthreads();
    }
    if (j == 0) {
        atomicAdd(out, red[0] * (1.0f / ((float)BIMG * S_SEG * S_SEG)));
    }
}

extern "C" void kernel_launch(void* const* d_in, const int* in_sizes, int n_in,
                              void* d_out, int out_size, void* d_ws, size_t ws_size,
                              hipStream_t stream) {
    const float* inp  = (const float*)d_in[0];   // [8,64,256,256] f32
    const float* feat = (const float*)d_in[1];   // [8,64,64,64]   f32
    const int*   sp   = (const int*)d_in[2];     // [8,1,256,256]  int
    // d_in[3] = num, always == S, unused

    float* out = (float*)d_out;
    float* ws  = (float*)d_ws;
    float* fm   = ws + OFF_FEATMEAN;
    float* sum1 = ws + OFF_SUM1;
    float* sum2 = ws + OFF_SUM2;
    float* cnt  = ws + OFF_CNT;
    float* m1   = ws + OFF_M1;
    float* m2   = ws + OFF_M2;

    k_zero<<<(3 * BIMG * S_SEG + 255) / 256, 256, 0, stream>>>(ws, out);
    k_featmean<<<(BIMG * FHW) / 256, 256, 0, stream>>>(feat, fm);
    k_main<<<BIMG * 128, 256, 0, stream>>>(inp, sp, fm, sum1, sum2, cnt);
    k_means<<<(BIMG * S_SEG) / 256, 256, 0, stream>>>(sum1, sum2, cnt, m1, m2);
    k_loss<<<BIMG, 256, 0, stream>>>(m1, m2, out);
}